// HeteroGraphEncoder_53412213293745
// MI455X (gfx1250) — compile-verified
//
#include <hip/hip_runtime.h>

typedef __attribute__((ext_vector_type(2))) float v2f;
typedef __attribute__((ext_vector_type(8))) float v8f;

#define HDIM 128

// ---------------- zero fill ----------------
__global__ void zero_f32_kernel(float* __restrict__ p, long long n) {
  long long i = (long long)blockIdx.x * blockDim.x + threadIdx.x;
  long long stride = (long long)gridDim.x * blockDim.x;
  for (; i < n; i += stride) p[i] = 0.0f;
}

// ---------------- segment-sum scatter: one wave per edge ----------------
// Coalesced 512B row read per wave; contiguous non-returning f32 atomics into
// an L2-resident msg buffer. This is the bandwidth/atomic-bound hot loop.
__global__ __launch_bounds__(256) void scatter_kernel(
    const float* __restrict__ h_src, const int* __restrict__ ei, int E,
    float* __restrict__ msg, float* __restrict__ cnt) {
  int wave = threadIdx.x >> 5;
  int lane = threadIdx.x & 31;
  long long e = (long long)blockIdx.x * 8 + wave;
  if (e >= E) return;
  int src = ei[e];
  int dst = ei[(long long)E + e];
  const float4* srow = (const float4*)(h_src + (long long)src * HDIM);
  float4 v = srow[lane];
  float* drow = msg + (long long)dst * HDIM + lane * 4;
  atomicAdd(drow + 0, v.x);
  atomicAdd(drow + 1, v.y);
  atomicAdd(drow + 2, v.z);
  atomicAdd(drow + 3, v.w);
  if (lane == 0) atomicAdd(cnt + dst, 1.0f);
}

// ---------------- fused SAGE GEMM via V_WMMA_F32_16X16X4_F32 ----------------
// out[M x 128] = init + (A1 * rowscale(cnt1)) @ W1[K1 x 128] + A2 @ W2[K2 x 128]
// init = bias broadcast (accsrc==null) or accsrc tile (accumulate across edge types).
// One wave computes one 16x16 tile; block = 8 waves = 16 rows x 128 cols.
// The 16xK A tile is staged (scaled) into LDS once per block, coalesced; the
// per-lane WMMA A fragments then come from conflict-free ds_load_b64.
__global__ __launch_bounds__(256) void gemm_kernel(
    const float* __restrict__ A1, const float* __restrict__ W1,
    const float* __restrict__ cnt1, int K1,
    const float* __restrict__ A2, const float* __restrict__ W2, int K2,
    const float* __restrict__ bias, const float* __restrict__ accsrc,
    float* __restrict__ out) {
  __shared__ float sInv[16];
  __shared__ float As[16 * (HDIM + 4)];  // +4 pad: lane m -> banks {4m+2h,4m+2h+1}

  const int t    = threadIdx.x;
  const int lane = t & 31;
  const int wv   = t >> 5;
  const int mod  = lane & 15;
  const int half = lane >> 4;
  const int m0   = blockIdx.x * 16;
  const int n0   = wv * 16;

  v8f c;
  if (accsrc) {
#pragma unroll
    for (int v = 0; v < 8; ++v)
      c[v] = accsrc[(long long)(m0 + v + 8 * half) * HDIM + n0 + mod];
  } else {
    float bv = bias[n0 + mod];
#pragma unroll
    for (int v = 0; v < 8; ++v) c[v] = bv;
  }

  // ---- stage term-1 A tile into LDS with mean scaling folded in ----
  if (t < 16) sInv[t] = cnt1 ? (1.0f / fmaxf(cnt1[m0 + t], 1.0f)) : 1.0f;
  __syncthreads();
  {
    const int ksh = 31 - __builtin_clz(K1);  // K1 in {32,64,128}
    const int lda = K1 + 4;
    for (int idx = t; idx < 16 * K1; idx += 256) {  // exact multiple of 256
      int row = idx >> ksh;
      int col = idx & (K1 - 1);
      As[row * lda + col] = A1[(long long)(m0 + row) * K1 + col] * sInv[row];
    }
  }
  __syncthreads();
  // ---- term 1 WMMA loop (A from LDS, B from global/WGP$) ----
  {
    const float* ap = &As[mod * (K1 + 4) + 2 * half];    // ds_load_b64 per step
    const float* bp = W1 + (2 * half) * HDIM + n0 + mod; // B[K=2h+v][N=mod]
    for (int k0 = 0; k0 < K1; k0 += 4) {
      v2f a, b;
      a[0] = ap[0];
      a[1] = ap[1];
      b[0] = bp[0];
      b[1] = bp[HDIM];
      c = __builtin_amdgcn_wmma_f32_16x16x4_f32(false, a, false, b, (short)0, c,
                                                false, false);
      ap += 4;
      bp += 4 * HDIM;
    }
  }
  // ---- term 2 (root transform, no scaling); A2 is uniform -> no divergence ----
  if (A2) {
    __syncthreads();  // all waves done reading term-1 tile
    const int ksh = 31 - __builtin_clz(K2);
    const int lda = K2 + 4;
    for (int idx = t; idx < 16 * K2; idx += 256) {
      int row = idx >> ksh;
      int col = idx & (K2 - 1);
      As[row * lda + col] = A2[(long long)(m0 + row) * K2 + col];
    }
    __syncthreads();
    const float* ap = &As[mod * (K2 + 4) + 2 * half];
    const float* bp = W2 + (2 * half) * HDIM + n0 + mod;
    for (int k0 = 0; k0 < K2; k0 += 4) {
      v2f a, b;
      a[0] = ap[0];
      a[1] = ap[1];
      b[0] = bp[0];
      b[1] = bp[HDIM];
      c = __builtin_amdgcn_wmma_f32_16x16x4_f32(false, a, false, b, (short)0, c,
                                                false, false);
      ap += 4;
      bp += 4 * HDIM;
    }
  }
#pragma unroll
  for (int v = 0; v < 8; ++v)
    out[(long long)(m0 + v + 8 * half) * HDIM + n0 + mod] = c[v];
}

// ---------------- column-wise BN statistics (sum, sumsq per feature) ----------------
__global__ __launch_bounds__(128) void bn_stats_kernel(
    const float* __restrict__ o, int M, float* __restrict__ stats) {
  int j = threadIdx.x;  // feature index 0..127
  float s = 0.0f, s2 = 0.0f;
  for (long long r = blockIdx.x; r < M; r += gridDim.x) {
    float x = o[r * HDIM + j];
    s += x;
    s2 += x * x;
  }
  atomicAdd(&stats[j], s);
  atomicAdd(&stats[HDIM + j], s2);
}

// ---------------- BN normalize + affine + ReLU ----------------
__global__ void bn_apply_kernel(const float* __restrict__ o,
                                const float* __restrict__ stats,
                                const float* __restrict__ gamma,
                                const float* __restrict__ beta,
                                float* __restrict__ h, long long M) {
  long long n = M * HDIM;
  long long i = (long long)blockIdx.x * blockDim.x + threadIdx.x;
  long long stride = (long long)gridDim.x * blockDim.x;
  float invM = 1.0f / (float)M;
  for (; i < n; i += stride) {
    int j = (int)(i & (HDIM - 1));
    float mean = stats[j] * invM;
    float var = stats[HDIM + j] * invM - mean * mean;
    float y = (o[i] - mean) * rsqrtf(var + 1e-5f) * gamma[j] + beta[j];
    h[i] = fmaxf(y, 0.0f);
  }
}

extern "C" void kernel_launch(void* const* d_in, const int* in_sizes, int n_in,
                              void* d_out, int out_size, void* d_ws, size_t ws_size,
                              hipStream_t stream) {
  (void)n_in; (void)out_size; (void)ws_size;
  const float* x_host     = (const float*)d_in[0];
  const float* x_proc     = (const float*)d_in[1];
  const int*   ei_hh      = (const int*)d_in[2];
  const int*   ei_ph      = (const int*)d_in[3];
  const int*   ei_hp      = (const int*)d_in[4];
  const float* Wp_host    = (const float*)d_in[5];
  const float* bp_host    = (const float*)d_in[6];
  const float* Wp_proc    = (const float*)d_in[7];
  const float* bp_proc    = (const float*)d_in[8];
  const float* Wl_hh      = (const float*)d_in[9];
  const float* bl_hh      = (const float*)d_in[10];
  const float* Wr_hh      = (const float*)d_in[11];
  const float* Wl_ph      = (const float*)d_in[12];
  const float* bl_ph      = (const float*)d_in[13];
  const float* Wr_ph      = (const float*)d_in[14];
  const float* Wl_hp      = (const float*)d_in[15];
  const float* bl_hp      = (const float*)d_in[16];
  const float* Wr_hp      = (const float*)d_in[17];
  const float* gamma_host = (const float*)d_in[18];
  const float* beta_host  = (const float*)d_in[19];
  const float* gamma_proc = (const float*)d_in[20];
  const float* beta_proc  = (const float*)d_in[21];

  const int NH = in_sizes[0] / 64;   // 100000
  const int NP = in_sizes[1] / 32;   // 200000
  const int E  = in_sizes[2] / 2;    // 1600000
  const int L  = 2;

  size_t off = 0;
  auto alloc = [&](long long nfloats) -> float* {
    float* p = (float*)((char*)d_ws + off);
    off += (size_t)nfloats * sizeof(float);
    off = (off + 255) & ~(size_t)255;
    return p;
  };
  float* h_host = alloc((long long)NH * HDIM);
  float* h_proc = alloc((long long)NP * HDIM);
  float* o_host = alloc((long long)NH * HDIM);
  float* o_proc = alloc((long long)NP * HDIM);
  float* msg    = alloc((long long)NP * HDIM);  // reused for all edge types
  float* cnt    = alloc(NP);
  float* stats  = alloc(2 * HDIM);
  float* outf   = (float*)d_out;

  // Input projections
  gemm_kernel<<<NH / 16, 256, 0, stream>>>(x_host, Wp_host, nullptr, 64,
                                           nullptr, nullptr, 0, bp_host, nullptr, h_host);
  gemm_kernel<<<NP / 16, 256, 0, stream>>>(x_proc, Wp_proc, nullptr, 32,
                                           nullptr, nullptr, 0, bp_proc, nullptr, h_proc);

  const int scatterBlocks = (E + 7) / 8;

  for (int l = 0; l < L; ++l) {
    const float* Wlhh = Wl_hh + (long long)l * HDIM * HDIM;
    const float* Wrhh = Wr_hh + (long long)l * HDIM * HDIM;
    const float* blhh = bl_hh + l * HDIM;
    const float* Wlph = Wl_ph + (long long)l * HDIM * HDIM;
    const float* Wrph = Wr_ph + (long long)l * HDIM * HDIM;
    const float* blph = bl_ph + l * HDIM;
    const float* Wlhp = Wl_hp + (long long)l * HDIM * HDIM;
    const float* Wrhp = Wr_hp + (long long)l * HDIM * HDIM;
    const float* blhp = bl_hp + l * HDIM;

    // o_host: host<-host edges
    zero_f32_kernel<<<2048, 256, 0, stream>>>(msg, (long long)NH * HDIM);
    zero_f32_kernel<<<256, 256, 0, stream>>>(cnt, NH);
    scatter_kernel<<<scatterBlocks, 256, 0, stream>>>(h_host, ei_hh, E, msg, cnt);
    gemm_kernel<<<NH / 16, 256, 0, stream>>>(msg, Wlhh, cnt, HDIM,
                                             h_host, Wrhh, HDIM, blhh, nullptr, o_host);
    // o_host += proc->host edges (HeteroConv aggr='sum')
    zero_f32_kernel<<<2048, 256, 0, stream>>>(msg, (long long)NH * HDIM);
    zero_f32_kernel<<<256, 256, 0, stream>>>(cnt, NH);
    scatter_kernel<<<scatterBlocks, 256, 0, stream>>>(h_proc, ei_ph, E, msg, cnt);
    gemm_kernel<<<NH / 16, 256, 0, stream>>>(msg, Wlph, cnt, HDIM,
                                             h_host, Wrph, HDIM, blph, o_host, o_host);
    // o_proc: host->proc edges (must run before BN overwrites h_host)
    zero_f32_kernel<<<2048, 256, 0, stream>>>(msg, (long long)NP * HDIM);
    zero_f32_kernel<<<256, 256, 0, stream>>>(cnt, NP);
    scatter_kernel<<<scatterBlocks, 256, 0, stream>>>(h_host, ei_hp, E, msg, cnt);
    gemm_kernel<<<NP / 16, 256, 0, stream>>>(msg, Wlhp, cnt, HDIM,
                                             h_proc, Wrhp, HDIM, blhp, nullptr, o_proc);

    // BatchNorm + ReLU; last layer writes into d_out in concat order
    float* h_host_dst = (l == L - 1) ? outf : h_host;
    float* h_proc_dst = (l == L - 1) ? outf + (long long)NH * HDIM : h_proc;

    zero_f32_kernel<<<1, 256, 0, stream>>>(stats, 2 * HDIM);
    bn_stats_kernel<<<1024, 128, 0, stream>>>(o_host, NH, stats);
    bn_apply_kernel<<<2048, 256, 0, stream>>>(o_host, stats, gamma_host + l * HDIM,
                                              beta_host + l * HDIM, h_host_dst, NH);

    zero_f32_kernel<<<1, 256, 0, stream>>>(stats, 2 * HDIM);
    bn_stats_kernel<<<1024, 128, 0, stream>>>(o_proc, NP, stats);
    bn_apply_kernel<<<2048, 256, 0, stream>>>(o_proc, stats, gamma_proc + l * HDIM,
                                              beta_proc + l * HDIM, h_proc_dst, NP);
  }
}